// WarpingLayer_Flow_20555713478763
// MI455X (gfx1250) — compile-verified
//
#include <hip/hip_runtime.h>
#include <hip/hip_bf16.h>

// Bilinear flow-warp (grid_sample, padding='zeros', align_corners=False) + validity mask.
// x: [B,64,256,512] f32, flow: [B,2,256,512] f32 -> out: [B,64,256,512] f32.
// Memory-bound: one thread per (b,h,w) pixel; weight/address math amortized over C channels.
// Compile-time dims -> pixel decomposition is shifts/ands, channel strides are immediates.

namespace {
constexpr int W  = 512;
constexpr int H  = 256;
constexpr int C  = 64;
constexpr int HW = H * W;          // 131072
constexpr int LOG2_W  = 9;
constexpr int LOG2_HW = 17;
}

__global__ __launch_bounds__(256) void warp_flow_kernel(
    const float* __restrict__ x, const float* __restrict__ flow,
    float* __restrict__ out, int npix)
{
    int idx = blockIdx.x * blockDim.x + threadIdx.x;
    if (idx >= npix) return;

    int w = idx & (W - 1);
    int h = (idx >> LOG2_W) & (H - 1);
    int b = idx >> LOG2_HW;

    // flow components (coalesced along w)
    const float* fl = flow + ((size_t)b * 2) * HW + (size_t)(h << LOG2_W) + w;
    float fx = fl[0];
    float fy = fl[HW];

    // base identity grid (linspace -1..1) + flow scaled by 2/(dim-1)
    constexpr float invWm1 = 1.0f / (float)(W - 1);
    constexpr float invHm1 = 1.0f / (float)(H - 1);
    float gx = -1.0f + 2.0f * (float)w * invWm1 + fx * (2.0f * invWm1);
    float gy = -1.0f + 2.0f * (float)h * invHm1 + fy * (2.0f * invHm1);

    // unnormalize (align_corners=False)
    float ix = ((gx + 1.0f) * (float)W - 1.0f) * 0.5f;
    float iy = ((gy + 1.0f) * (float)H - 1.0f) * 0.5f;

    float ix0f = floorf(ix);
    float iy0f = floorf(iy);
    float wx1 = ix - ix0f, wy1 = iy - iy0f;
    float wx0 = 1.0f - wx1, wy0 = 1.0f - wy1;

    int ix0 = (int)ix0f, iy0 = (int)iy0f;
    int ix1 = ix0 + 1,   iy1 = iy0 + 1;

    // per-corner validity folded into the weights (zeros padding)
    float vx0 = (ix0 >= 0 && ix0 < W) ? 1.0f : 0.0f;
    float vx1 = (ix1 >= 0 && ix1 < W) ? 1.0f : 0.0f;
    float vy0 = (iy0 >= 0 && iy0 < H) ? 1.0f : 0.0f;
    float vy1 = (iy1 >= 0 && iy1 < H) ? 1.0f : 0.0f;

    float w00 = wy0 * wx0 * (vy0 * vx0);
    float w01 = wy0 * wx1 * (vy0 * vx1);
    float w10 = wy1 * wx0 * (vy1 * vx0);
    float w11 = wy1 * wx1 * (vy1 * vx1);

    // mask = grid_sample(ones) >= 1.0 ; same summation order as reference
    float msum = ((w00 + w01) + w10) + w11;
    float maskf = (msum >= 1.0f) ? 1.0f : 0.0f;
    // mask is exactly 0 or 1, so folding it into the weights distributes exactly
    w00 *= maskf; w01 *= maskf; w10 *= maskf; w11 *= maskf;

    // clamped gather offsets (invalid corners carry weight 0)
    int xc0 = min(max(ix0, 0), W - 1);
    int xc1 = min(max(ix1, 0), W - 1);
    int yc0 = min(max(iy0, 0), H - 1);
    int yc1 = min(max(iy1, 0), H - 1);

    int o00 = (yc0 << LOG2_W) + xc0;
    int o01 = (yc0 << LOG2_W) + xc1;
    int o10 = (yc1 << LOG2_W) + xc0;
    int o11 = (yc1 << LOG2_W) + xc1;

    const float* xb = x   + (size_t)b * C * HW;
    float*       ob = out + (size_t)b * C * HW + (size_t)(h << LOG2_W) + w;

    constexpr int PF = 8;  // channel-plane prefetch distance
    #pragma unroll 8
    for (int c = 0; c < C; ++c) {
        const float* xp = xb + (size_t)c * HW;
        if ((c & 7) == 0 && (c + PF) < C) {
            // gfx1250 global_prefetch_b8: pull both tap-row cachelines PF planes ahead
            __builtin_prefetch(xp + (size_t)PF * HW + o00, 0, 0);
            __builtin_prefetch(xp + (size_t)PF * HW + o10, 0, 0);
        }
        float r = xp[o00] * w00 + xp[o01] * w01 + xp[o10] * w10 + xp[o11] * w11;
        ob[(size_t)c * HW] = r;
    }
}

extern "C" void kernel_launch(void* const* d_in, const int* in_sizes, int n_in,
                              void* d_out, int out_size, void* d_ws, size_t ws_size,
                              hipStream_t stream) {
    const float* x    = (const float*)d_in[0];
    const float* flow = (const float*)d_in[1];
    float*       out  = (float*)d_out;

    int B = in_sizes[1] / (2 * HW);          // flow is [B,2,H,W]
    int npix = B * HW;                        // 1,048,576 for B=8
    const int threads = 256;                  // 8 wave32s per block
    int blocks = (npix + threads - 1) / threads;
    warp_flow_kernel<<<blocks, threads, 0, stream>>>(x, flow, out, npix);
}